// ComplexNet_89739046683231
// MI455X (gfx1250) — compile-verified
//
#include <hip/hip_runtime.h>

typedef __attribute__((ext_vector_type(2))) float v2f;
typedef __attribute__((ext_vector_type(8))) float v8f;

#define N_FEAT 10
#define N_CLS  2
// B tile padded to K=12 (3 WMMA k-steps of 4), N=16
#define KPAD   12
// Tiles (16 rows each) processed by one wave: amortizes B-load + setup.
#define TILES_PER_WAVE 8
#define ROWS_PER_WAVE  (16 * TILES_PER_WAVE)       // 128
#define ROWS_PER_BLOCK (8 * ROWS_PER_WAVE)         // 1024 (8 waves / block)

// ---------------------------------------------------------------------------
// Kernel 1: build the padded 12x16 coefficient tile Bpad[a][k] = M_r[k][a]
//   M_r[k,a] = sum_{i,j} (pr_j - pi_j) * (pr_i*A_r[k,i,j,a] - pi_i*A_i[k,i,j,a])
// Rows a>=10 and cols k>=2 are zero. Tiny one-block kernel.
// ---------------------------------------------------------------------------
__global__ void build_coeff_kernel(const float* __restrict__ psi_r,
                                   const float* __restrict__ psi_i,
                                   const float* __restrict__ A_r,
                                   const float* __restrict__ A_i,
                                   float* __restrict__ Bpad) {
    int t = threadIdx.x;
    if (t >= KPAD * 16) return;
    int a = t >> 4;   // K dimension (feature index), 0..11
    int k = t & 15;   // N dimension (class index),   0..15
    float val = 0.0f;
    if (a < N_FEAT && k < N_CLS) {
        for (int i = 0; i < N_FEAT; ++i) {
            float pri = psi_r[i];
            float pii = psi_i[i];
            for (int j = 0; j < N_FEAT; ++j) {
                float w = psi_r[j] - psi_i[j];
                int idx = ((k * N_FEAT + i) * N_FEAT + j) * N_FEAT + a;
                val += w * (pri * A_r[idx] - pii * A_i[idx]);
            }
        }
    }
    Bpad[t] = val;
}

// ---------------------------------------------------------------------------
// Kernel 2: out[T,2] = x[T,10] @ M_r.T via V_WMMA_F32_16X16X4_F32.
// Each wave streams 8 consecutive 16-row tiles; B fragments loaded once.
// Per tile: 3x global_load_b64 (coalesced 640B/tile) + 3x WMMA + stores.
// ---------------------------------------------------------------------------
__global__ void __launch_bounds__(256) gemv_wmma_kernel(
    const float* __restrict__ X,
    const float* __restrict__ Bpad,
    float* __restrict__ Out,
    int T) {
    const int lane = threadIdx.x & 31;
    const int wave = threadIdx.x >> 5;
    const int hi   = lane >> 4;   // 0: lanes 0-15, 1: lanes 16-31
    const int n    = lane & 15;   // N (class) column / A-tile row index

    const long long waveBase =
        (long long)blockIdx.x * ROWS_PER_BLOCK + (long long)wave * ROWS_PER_WAVE;
    if (waveBase >= T) return;    // wave-uniform exit, EXEC stays all-ones

    // --- B fragments: loaded once per wave, live across the whole tile loop.
    // VGPR v of chunk c holds Bpad[(4c + v + 2*hi)][n] (zero-padded rows/cols).
    v2f bf[3];
#pragma unroll
    for (int c = 0; c < 3; ++c) {
        bf[c].x = Bpad[(4 * c + 0 + hi * 2) * 16 + n];
        bf[c].y = Bpad[(4 * c + 1 + hi * 2) * 16 + n];
    }

#pragma unroll
    for (int t = 0; t < TILES_PER_WAVE; ++t) {
        const long long rowbase = waveBase + t * 16;
        if (rowbase >= T) break;  // wave-uniform, EXEC stays all-ones

        // A-fragment source row for this lane (clamped for the ragged last
        // tile; duplicated rows are computed but their stores are masked).
        long long rowA = rowbase + n;
        if (rowA > (long long)T - 1) rowA = (long long)T - 1;
        const float* xrow = X + rowA * N_FEAT;

        // A chunk loads: lane holds x[row][kb], x[row][kb+1], kb = k0 + 2*hi.
        // Chunk 2 uses kb=8 for both halves (k=10,11 rows of B are zero), so
        // every load stays in-bounds and 8B-aligned.
        v2f a0 = *(const v2f*)(xrow + 0 + hi * 2);
        v2f a1 = *(const v2f*)(xrow + 4 + hi * 2);
        v2f a2 = *(const v2f*)(xrow + 8);

        v8f c = {};
        // (neg_a, A, neg_b, B, c_mod, C, reuse_a, reuse_b)
        c = __builtin_amdgcn_wmma_f32_16x16x4_f32(false, a0, false, bf[0],
                                                  (short)0, c, false, false);
        c = __builtin_amdgcn_wmma_f32_16x16x4_f32(false, a1, false, bf[1],
                                                  (short)0, c, false, false);
        c = __builtin_amdgcn_wmma_f32_16x16x4_f32(false, a2, false, bf[2],
                                                  (short)0, c, false, false);

        // D layout: VGPR v = row (v + 8*hi), col n. Only cols 0..1 are real.
        if (n < N_CLS) {
#pragma unroll
            for (int v = 0; v < 8; ++v) {
                long long row = rowbase + hi * 8 + v;
                if (row < T) Out[row * N_CLS + n] = c[v];
            }
        }
    }
}

// ---------------------------------------------------------------------------
extern "C" void kernel_launch(void* const* d_in, const int* in_sizes, int n_in,
                              void* d_out, int out_size, void* d_ws, size_t ws_size,
                              hipStream_t stream) {
    const float* x     = (const float*)d_in[0];   // [T, 10]
    const float* psi_r = (const float*)d_in[1];   // [10]
    const float* psi_i = (const float*)d_in[2];   // [10]
    const float* A_r   = (const float*)d_in[3];   // [2,10,10,10]
    const float* A_i   = (const float*)d_in[4];   // [2,10,10,10]
    float* out  = (float*)d_out;                  // [T, 2]
    float* Bpad = (float*)d_ws;                   // 12*16 floats = 768 B

    const int T = in_sizes[0] / N_FEAT;

    build_coeff_kernel<<<1, 256, 0, stream>>>(psi_r, psi_i, A_r, A_i, Bpad);

    const int grid = (T + ROWS_PER_BLOCK - 1) / ROWS_PER_BLOCK;
    gemv_wmma_kernel<<<grid, 256, 0, stream>>>(x, Bpad, out, T);
}